// VladCore_14190571946190
// MI455X (gfx1250) — compile-verified
//
#include <hip/hip_runtime.h>
#include <hip/hip_bf16.h>

typedef float v2f __attribute__((ext_vector_type(2)));
typedef float v8f __attribute__((ext_vector_type(8)));

#define TILE_K 4      // K per V_WMMA_F32_16X16X4_F32
#define UNROLL 4      // 4 WMMAs / 16 n-elements per loop iteration
#define CHUNK  4096   // split-K chunk of the n dimension per workgroup

// V = X (D x Nn) * A^T (Nn x Kc), Kc == 64, split-K with f32 atomics.
// One block = 4 waves; wave w computes 16x16 output tile (mtile, w).
__global__ __launch_bounds__(128)
void vlad_wmma_f32_kernel(const float* __restrict__ X,
                          const float* __restrict__ A,
                          float* __restrict__ out,
                          long long Nn)
{
    const int lane = threadIdx.x & 31;
    const int wave = threadIdx.x >> 5;     // 0..3 -> which 16-wide cluster tile
    const int half = lane >> 4;            // lane half selects K-pair {0,1} vs {2,3}
    const int l16  = lane & 15;

    const int mtile = blockIdx.y;                      // 0..D/16-1
    const long long n0 = (long long)blockIdx.x * CHUNK;

    const int m    = (mtile << 4) + l16;   // X row this lane supplies (A-fragment)
    const int kcol = (wave  << 4) + l16;   // cluster column this lane supplies (B-fragment)

    // A-fragment: lane loads X[m][n + 2*half .. +1]  (contiguous -> b64)
    // B-fragment: lane loads A[kcol][n + 2*half .. +1] (contiguous -> b64)
    const float* xp = X + (long long)m    * Nn + n0 + 2 * half;
    const float* ap = A + (long long)kcol * Nn + n0 + 2 * half;

    v8f acc0 = {0.f, 0.f, 0.f, 0.f, 0.f, 0.f, 0.f, 0.f};
    v8f acc1 = {0.f, 0.f, 0.f, 0.f, 0.f, 0.f, 0.f, 0.f};

    for (int i = 0; i < CHUNK; i += TILE_K * UNROLL) {
        // stream ahead (gfx1250 global_prefetch_b8); A stays L2-resident (192 MB L2)
        __builtin_prefetch(xp + i + 128, 0, 1);
        __builtin_prefetch(ap + i + 128, 0, 1);
#pragma unroll
        for (int u = 0; u < UNROLL; ++u) {
            v2f av = *(const v2f*)(xp + i + TILE_K * u);
            v2f bv = *(const v2f*)(ap + i + TILE_K * u);
            // 8 args: (neg_a, A, neg_b, B, c_mod, C, reuse_a, reuse_b)
            if (u & 1)
                acc1 = __builtin_amdgcn_wmma_f32_16x16x4_f32(
                    false, av, false, bv, (short)0, acc1, false, false);
            else
                acc0 = __builtin_amdgcn_wmma_f32_16x16x4_f32(
                    false, av, false, bv, (short)0, acc0, false, false);
        }
    }

    // C/D layout: VGPR g holds (M = g, N = lane) for lanes 0-15,
    //             (M = g+8, N = lane-16) for lanes 16-31.
    // Output V is (D x 64) row-major: out[mrow*64 + kcol].
#pragma unroll
    for (int g = 0; g < 8; ++g) {
        const int mrow = (mtile << 4) + g + 8 * half;
        atomicAdd(out + (long long)mrow * 64 + kcol, acc0[g] + acc1[g]);
    }
}

__global__ void vlad_zero_kernel(float* __restrict__ p, int n)
{
    int i = blockIdx.x * 256 + threadIdx.x;
    if (i < n) p[i] = 0.0f;
}

extern "C" void kernel_launch(void* const* d_in, const int* in_sizes, int n_in,
                              void* d_out, int out_size, void* d_ws, size_t ws_size,
                              hipStream_t stream) {
    (void)n_in; (void)d_ws; (void)ws_size;

    const float* x = (const float*)d_in[0];   // (D, Nn) fp32
    const float* a = (const float*)d_in[1];   // (1, 64, H, W) -> flat (64, Nn) fp32
    float* out = (float*)d_out;               // (D, 64) fp32

    const int Kc = 64;
    const long long Nn = (long long)in_sizes[1] / Kc;   // 262144
    const int D = (int)((long long)in_sizes[0] / Nn);   // 512

    // Zero the accumulation target (harness poisons d_out).
    vlad_zero_kernel<<<(out_size + 255) / 256, 256, 0, stream>>>(out, out_size);

    dim3 block(128, 1, 1);                               // 4 waves (wave32)
    dim3 grid((unsigned)(Nn / CHUNK), (unsigned)(D / 16), 1);
    vlad_wmma_f32_kernel<<<grid, block, 0, stream>>>(x, a, out, Nn);
}